// Qu_mutual_6794638263032
// MI455X (gfx1250) — compile-verified
//
#include <hip/hip_runtime.h>
#include <math.h>

// ---------------------------------------------------------------------------
// Qu_mutual on gfx1250: complex64 unitary conjugation via fp32 WMMA (16x16x4)
// All GEMM B-operands consumed transposed => contiguous b64 loads everywhere.
// ---------------------------------------------------------------------------

typedef __attribute__((ext_vector_type(2))) float v2f;
typedef __attribute__((ext_vector_type(8))) float v8f;

#define QDIM 1024   // 2^10
#define NQ   10

static __device__ __forceinline__ v8f wmma_f32(v2f a, v2f b, v8f c) {
    // 8-arg form: (neg_a, A, neg_b, B, c_mod, C, reuse_a, reuse_b)
    return __builtin_amdgcn_wmma_f32_16x16x4_f32(
        false, a, false, b, (short)0, c, false, false);
}

// CNOT chain as ket permutation: b1^=b0; b2^=b1; ... b9^=b8; b0^=b9 (sequential)
static __device__ __forceinline__ unsigned perm10(unsigned x) {
    unsigned b[NQ];
#pragma unroll
    for (int q = 0; q < NQ; ++q) b[q] = (x >> (9 - q)) & 1u;
#pragma unroll
    for (int q = 0; q < NQ - 1; ++q) b[q + 1] ^= b[q];
    b[0] ^= b[9];
    unsigned y = 0;
#pragma unroll
    for (int q = 0; q < NQ; ++q) y |= b[q] << (9 - q);
    return y;
}

// ---- Kernel 1: per-qubit 2x2 gate matrices M = Rz * Ry * Rx  (20 of them) --
__global__ void k_build_gates(const float* __restrict__ w,
                              float* __restrict__ gates) {
    int t = threadIdx.x;
    if (t >= 2 * NQ) return;
    int layer = t / NQ, q = t % NQ;
    const float WM = 0.63245553203367586640f;  // sqrt(2/5) = he_std*lrmul
    float ax = w[6 * q + 3 * layer + 0] * WM * 0.5f;
    float ay = w[6 * q + 3 * layer + 1] * WM * 0.5f;
    float az = w[6 * q + 3 * layer + 2] * WM * 0.5f;
    float cx = cosf(ax), sx = sinf(ax);
    float cy = cosf(ay), sy = sinf(ay);
    float cz = cosf(az), sz = sinf(az);
    // T = Ry @ Rx  (Rx = [[cx,-i sx],[-i sx,cx]], Ry = [[cy,-sy],[sy,cy]])
    float T00r = cy * cx,  T00i =  sy * sx;
    float T01r = -sy * cx, T01i = -cy * sx;
    float T10r = sy * cx,  T10i = -cy * sx;
    float T11r = cy * cx,  T11i = -sy * sx;
    // M = Rz @ T : row0 *= (cz - i sz), row1 *= (cz + i sz)
    float* g = gates + t * 8;
    g[0] = T00r * cz + T00i * sz;  g[1] = T00i * cz - T00r * sz;
    g[2] = T01r * cz + T01i * sz;  g[3] = T01i * cz - T01r * sz;
    g[4] = T10r * cz - T10i * sz;  g[5] = T10i * cz + T10r * sz;
    g[6] = T11r * cz - T11i * sz;  g[7] = T11i * cz + T11r * sz;
}

// ---- Kernel 2: L2[r,c] = prod_q M2_q[rq,cq] (row-major, A-side);
//                PL1^T[c, perm(r)] = prod_q M1_q[rq,cq] (transposed, B-side) --
__global__ void k_build_layers(const float* __restrict__ gates,
                               float* __restrict__ L2re, float* __restrict__ L2im,
                               float* __restrict__ PL1Tre, float* __restrict__ PL1Tim) {
    __shared__ float g[160];
    int tid = threadIdx.x;
    if (tid < 160) g[tid] = gates[tid];
    __syncthreads();
    int idx = blockIdx.x * blockDim.x + tid;
    int r = idx >> 10, c = idx & (QDIM - 1);
    float p1r = 1.f, p1i = 0.f, p2r = 1.f, p2i = 0.f;
#pragma unroll
    for (int q = 0; q < NQ; ++q) {
        int rb = (r >> (9 - q)) & 1;
        int cb = (c >> (9 - q)) & 1;
        int e  = (rb * 2 + cb) * 2;
        float g1r = g[q * 8 + e],      g1i = g[q * 8 + e + 1];
        float g2r = g[80 + q * 8 + e], g2i = g[80 + q * 8 + e + 1];
        float t1r = p1r * g1r - p1i * g1i, t1i = p1r * g1i + p1i * g1r;
        float t2r = p2r * g2r - p2i * g2i, t2i = p2r * g2i + p2i * g2r;
        p1r = t1r; p1i = t1i; p2r = t2r; p2i = t2i;
    }
    L2re[idx] = p2r;
    L2im[idx] = p2i;
    unsigned pr = perm10((unsigned)r);
    PL1Tre[c * QDIM + pr] = p1r;      // store transposed
    PL1Tim[c * QDIM + pr] = p1i;
}

// ---- Kernel 3: complex GEMM with transposed B:
//      SIGN=+1 : C = A @ B          where BTre/BTim hold B^T (row-major)
//      SIGN=-1 : C = A @ conj(B^T)  where BTre/BTim hold B itself (== U @ U†)
// B fragment: B[k,c] = BT[c,k]  => contiguous in k per lane.
template <int SIGN>
__global__ __launch_bounds__(32)
void k_gemm_bt(const float* __restrict__ Are, const float* __restrict__ Aim,
               const float* __restrict__ BTre, const float* __restrict__ BTim,
               float* __restrict__ Cre, float* __restrict__ Cim) {
    const int lane = threadIdx.x;
    const int m = lane & 15, hi = lane >> 4;
    const int row0 = blockIdx.y << 4, col0 = blockIdx.x << 4;
    const int c = col0 + m;
    const float* ar = Are + (row0 + m) * QDIM;
    const float* ai = Aim + (row0 + m) * QDIM;
    const float* br = BTre + c * QDIM;
    const float* bi = BTim + c * QDIM;
    v8f acc_rr = {}, acc_ii = {}, acc_ri = {}, acc_ir = {};
#pragma unroll 4
    for (int k0 = 0; k0 < QDIM; k0 += 4) {
        int k = k0 + hi * 2;
        v2f fAr, fAi, fBr, fBi;
        float2 a2r = *reinterpret_cast<const float2*>(ar + k);
        float2 a2i = *reinterpret_cast<const float2*>(ai + k);
        float2 b2r = *reinterpret_cast<const float2*>(br + k);
        float2 b2i = *reinterpret_cast<const float2*>(bi + k);
        fAr.x = a2r.x; fAr.y = a2r.y;
        fAi.x = a2i.x; fAi.y = a2i.y;
        fBr.x = b2r.x; fBr.y = b2r.y;
        if (SIGN > 0) { fBi.x =  b2i.x; fBi.y =  b2i.y; }
        else          { fBi.x = -b2i.x; fBi.y = -b2i.y; }   // conjugate
        acc_rr = wmma_f32(fAr, fBr, acc_rr);
        acc_ii = wmma_f32(fAi, fBi, acc_ii);
        acc_ri = wmma_f32(fAr, fBi, acc_ri);
        acc_ir = wmma_f32(fAi, fBr, acc_ir);
    }
#pragma unroll
    for (int v = 0; v < 8; ++v) {
        int rr = row0 + v + 8 * hi;
        Cre[rr * QDIM + c] = acc_rr[v] - acc_ii[v];
        Cim[rr * QDIM + c] = acc_ri[v] + acc_ir[v];
    }
}

// ---- Kernel 4: T = U @ kron(P,Q)  (B real, built on the fly from LDS) ------
__global__ __launch_bounds__(32)
void k_gemm_kron(const float* __restrict__ Ure, const float* __restrict__ Uim,
                 const float* __restrict__ P, const float* __restrict__ Q,
                 float* __restrict__ Tre, float* __restrict__ Tim) {
    __shared__ float sP[1024], sQ[1024];
    for (int i = threadIdx.x; i < 1024; i += 32) { sP[i] = P[i]; sQ[i] = Q[i]; }
    __syncthreads();
    const int lane = threadIdx.x;
    const int m = lane & 15, hi = lane >> 4;
    const int row0 = blockIdx.y << 4, col0 = blockIdx.x << 4;
    const int c = col0 + m, cHi = c >> 5, cLo = c & 31;
    const float* ur = Ure + (row0 + m) * QDIM;
    const float* ui = Uim + (row0 + m) * QDIM;
    v8f accr = {}, acci = {};
#pragma unroll 4
    for (int k0 = 0; k0 < QDIM; k0 += 4) {
        int k = k0 + hi * 2;
        v2f fAr, fAi, fB;
        float2 a2r = *reinterpret_cast<const float2*>(ur + k);
        float2 a2i = *reinterpret_cast<const float2*>(ui + k);
        fAr.x = a2r.x; fAr.y = a2r.y;
        fAi.x = a2i.x; fAi.y = a2i.y;
        fB.x = sP[(k >> 5) * 32 + cHi] * sQ[(k & 31) * 32 + cLo];
        int k1 = k + 1;
        fB.y = sP[(k1 >> 5) * 32 + cHi] * sQ[(k1 & 31) * 32 + cLo];
        accr = wmma_f32(fAr, fB, accr);
        acci = wmma_f32(fAi, fB, acci);
    }
#pragma unroll
    for (int v = 0; v < 8; ++v) {
        int rr = row0 + v + 8 * hi;
        Tre[rr * QDIM + c] = accr[v];
        Tim[rr * QDIM + c] = acci[v];
    }
}

// ---- Kernel 5: partial trace over 2nd subsystem -> interleaved complex64 ---
__global__ void k_ptrace(const float* __restrict__ Sre,
                         const float* __restrict__ Sim,
                         float* __restrict__ out) {
    int t = blockIdx.x * blockDim.x + threadIdx.x;  // 0..1023
    int i = t >> 5, j = t & 31;
    float re = 0.f, im = 0.f;
#pragma unroll 4
    for (int b = 0; b < 32; ++b) {
        int r = i * 32 + b, c = j * 32 + b;
        re += Sre[r * QDIM + c];
        im += Sim[r * QDIM + c];
    }
    out[2 * t + 0] = re;
    out[2 * t + 1] = im;
}

// ---------------------------------------------------------------------------
extern "C" void kernel_launch(void* const* d_in, const int* in_sizes, int n_in,
                              void* d_out, int out_size, void* d_ws, size_t ws_size,
                              hipStream_t stream) {
    const float* inA = (const float*)d_in[0];   // 32x32 f32
    const float* inB = (const float*)d_in[1];   // 32x32 f32
    const float* w   = (const float*)d_in[2];   // 60 f32
    float* ws = (float*)d_ws;
    const size_t M = (size_t)QDIM * QDIM;
    float* bAre = ws + 0 * M;  float* bAim = ws + 1 * M;
    float* bBre = ws + 2 * M;  float* bBim = ws + 3 * M;
    float* bCre = ws + 4 * M;  float* bCim = ws + 5 * M;
    float* gates = ws + 6 * M;                  // 160 floats
    float* out = (float*)d_out;

    dim3 g64(64, 64), b32(32, 1, 1);

    k_build_gates<<<dim3(1), dim3(32), 0, stream>>>(w, gates);
    // L2 -> bufA (row-major), (P*L1)^T -> bufB (transposed)
    k_build_layers<<<dim3(4096), dim3(256), 0, stream>>>(gates, bAre, bAim, bBre, bBim);
    // U = L2 @ (P*L1) -> bufC   (B supplied transposed, no conjugation)
    k_gemm_bt<+1><<<g64, b32, 0, stream>>>(bAre, bAim, bBre, bBim, bCre, bCim);

    // ---- pass AB: rho = kron(A,B) ----
    k_gemm_kron   <<<g64, b32, 0, stream>>>(bCre, bCim, inA, inB, bAre, bAim);  // T
    k_gemm_bt<-1> <<<g64, b32, 0, stream>>>(bAre, bAim, bCre, bCim, bBre, bBim); // S = T @ U†
    k_ptrace      <<<dim3(4), dim3(256), 0, stream>>>(bBre, bBim, out);

    // ---- pass BA: rho = kron(B,A) ----
    k_gemm_kron   <<<g64, b32, 0, stream>>>(bCre, bCim, inB, inA, bAre, bAim);
    k_gemm_bt<-1> <<<g64, b32, 0, stream>>>(bAre, bAim, bCre, bCim, bBre, bBim);
    k_ptrace      <<<dim3(4), dim3(256), 0, stream>>>(bBre, bBim, out + 2048);
}